// SensorResponse_26396869001941
// MI455X (gfx1250) — compile-verified
//
#include <hip/hip_runtime.h>

typedef __attribute__((ext_vector_type(2))) float v2f;
typedef __attribute__((ext_vector_type(8))) float v8f;

namespace {
constexpr int B_ = 8, N_ = 10000, T_ = 2000, S_ = 12, H_ = 128;
constexpr int WAVES = 4;            // waves per workgroup (128 threads)
constexpr int TILE_M = 16;          // points per wave tile
constexpr int ROWSTRIDE = H_ + 2;   // LDS row pad: float2 A-loads conflict-free
constexpr float INV_TWO_SIGMA = 5.0f;               // 1/(2*0.1)
constexpr float GNORM = 1.2615662610100802f;        // 0.3989422804/sqrt(0.1)
constexpr float RAD = 3.0f;                         // exp(-45) ~ 3e-20 cutoff
}

__global__ void zero_f32_kernel(float* __restrict__ p, int n) {
    int i = blockIdx.x * blockDim.x + threadIdx.x;
    if (i < n) p[i] = 0.0f;
}

__global__ void __launch_bounds__(WAVES * 32)
sensor_mlp_scatter(const float* __restrict__ X,    // (B*N, 3)
                   const float* __restrict__ mask, // (B*N)
                   const float* __restrict__ W1,   // (3, H)
                   const float* __restrict__ b1,   // (H)
                   const float* __restrict__ W2,   // (H, H)
                   const float* __restrict__ b2,   // (H)
                   const float* __restrict__ W3,   // (H, S)
                   const float* __restrict__ b3,   // (S)
                   float* __restrict__ out)        // (B, S, T)
{
    __shared__ float Hlds[WAVES][TILE_M][ROWSTRIDE];
    __shared__ float zbuf[WAVES][TILE_M];
    __shared__ float wbuf[WAVES][TILE_M];

    const int lane = threadIdx.x & 31;
    const int w    = threadIdx.x >> 5;
    const int n    = lane & 15;     // column / row-in-tile index
    const int h    = lane >> 4;     // lane half: holds K = {2h, 2h+1}
    const int tile = blockIdx.x * WAVES + w;
    const int base = tile * TILE_M; // first flattened point (b*N + i)

    const float hsel = (h == 0) ? 1.0f : 0.0f;   // zero-pad selector for K=3
    const float ssel = (n < S_) ? 1.0f : 0.0f;   // zero-pad selector for S->16
    const int   colS = (n < S_) ? n : 0;         // clamped, always in-bounds

    // ---- inputs + layer-1 A fragment (16x4 fp32, K=3 padded to 4) ------
    // All lanes load unconditionally (in-bounds), then select per lane-half.
    const int g = base + n;
    const float xx = X[3 * g + 0];
    const float xy = X[3 * g + 1];
    const float xz = X[3 * g + 2];
    v2f afrag;
    afrag.x = (h == 0) ? xx : xz;
    afrag.y = xy * hsel;            // K=1 for h==0, zero pad (K=3) for h==1
    if (h == 0) {
        zbuf[w][n] = xz;
        wbuf[w][n] = mask[g] * GNORM;
    }

    // ---- layer 1: H1 = relu(X @ W1 + b1); one K-step per 16-col tile ---
    #pragma unroll
    for (int j = 0; j < H_ / 16; ++j) {
        const int col = 16 * j + n;
        v2f bfrag;
        bfrag.x = W1[(2 * h) * H_ + col];        // row 0 (h=0) / row 2 (h=1)
        bfrag.y = W1[1 * H_ + col] * hsel;       // row 1 (h=0) / zero (h=1)
        v8f acc = {};
        acc = __builtin_amdgcn_wmma_f32_16x16x4_f32(false, afrag, false, bfrag,
                                                    (short)0, acc, false, false);
        const float bias = b1[col];
        #pragma unroll
        for (int r = 0; r < 8; ++r)
            Hlds[w][r + 8 * h][col] = fmaxf(acc[r] + bias, 0.0f);
    }

    // ---- layer 2: H2 = relu(H1 @ W2 + b2); H2 tiles held in VGPRs ------
    v8f h2[H_ / 16];
    #pragma unroll
    for (int j = 0; j < H_ / 16; ++j) {
        const int col = 16 * j + n;
        v8f acc = {};
        for (int k0 = 0; k0 < H_; k0 += 4) {
            const int k = k0 + 2 * h;
            v2f a = *(const v2f*)&Hlds[w][n][k];
            v2f b;
            b.x = W2[(k + 0) * H_ + col];
            b.y = W2[(k + 1) * H_ + col];
            acc = __builtin_amdgcn_wmma_f32_16x16x4_f32(false, a, false, b,
                                                        (short)0, acc, false, false);
        }
        const float bias = b2[col];
        #pragma unroll
        for (int r = 0; r < 8; ++r)
            acc[r] = fmaxf(acc[r] + bias, 0.0f);
        h2[j] = acc;
    }
    // overwrite per-wave LDS tile with H2 (no cross-wave sharing)
    #pragma unroll
    for (int j = 0; j < H_ / 16; ++j) {
        const int col = 16 * j + n;
        #pragma unroll
        for (int r = 0; r < 8; ++r)
            Hlds[w][r + 8 * h][col] = h2[j][r];
    }

    // ---- layer 3: R = H2 @ W3 + b3 (S=12 padded to 16 columns) ---------
    // Unconditional loads from a clamped column; zero the pad lanes by mul.
    v8f acc3 = {};
    for (int k0 = 0; k0 < H_; k0 += 4) {
        const int k = k0 + 2 * h;
        v2f a = *(const v2f*)&Hlds[w][n][k];
        v2f b;
        b.x = W3[(k + 0) * S_ + colS] * ssel;
        b.y = W3[(k + 1) * S_ + colS] * ssel;
        acc3 = __builtin_amdgcn_wmma_f32_16x16x4_f32(false, a, false, b,
                                                     (short)0, acc3, false, false);
    }

    // ---- Gaussian scatter: out[b][s][t] += cw * exp(-5 d^2) * resp -----
    if (n < S_) {
        const float bias3 = b3[n];
        #pragma unroll
        for (int r = 0; r < 8; ++r) {
            const int p  = r + 8 * h;          // point row in tile
            const int gp = base + p;           // flattened b*N + i
            const int bb = gp / N_;            // batch index
            const float z    = zbuf[w][p];
            const float cw   = wbuf[w][p];
            const float resp = acc3[r] + bias3;
            int t0 = (int)ceilf(z - RAD);  if (t0 < 0) t0 = 0;
            int t1 = (int)floorf(z + RAD); if (t1 > T_ - 1) t1 = T_ - 1;
            float* op = out + ((size_t)bb * S_ + n) * T_;
            for (int t = t0; t <= t1; ++t) {
                const float d   = (float)t - z;
                const float val = cw * __expf(-INV_TWO_SIGMA * d * d) * resp;
                __hip_atomic_fetch_add(&op[t], val, __ATOMIC_RELAXED,
                                       __HIP_MEMORY_SCOPE_AGENT);
            }
        }
    }
}

extern "C" void kernel_launch(void* const* d_in, const int* in_sizes, int n_in,
                              void* d_out, int out_size, void* d_ws, size_t ws_size,
                              hipStream_t stream) {
    (void)in_sizes; (void)n_in; (void)d_ws; (void)ws_size;
    const float* X    = (const float*)d_in[0];
    const float* mask = (const float*)d_in[1];
    const float* W1   = (const float*)d_in[2];
    const float* b1   = (const float*)d_in[3];
    const float* W2   = (const float*)d_in[4];
    const float* b2   = (const float*)d_in[5];
    const float* W3   = (const float*)d_in[6];
    const float* b3   = (const float*)d_in[7];
    float* out = (float*)d_out;

    // accumulation target must start at zero (harness poisons d_out)
    zero_f32_kernel<<<(out_size + 255) / 256, 256, 0, stream>>>(out, out_size);

    const int tiles  = (B_ * N_) / TILE_M;  // 5000 wave tiles
    const int blocks = tiles / WAVES;       // 1250 blocks of 128 threads
    sensor_mlp_scatter<<<blocks, WAVES * 32, 0, stream>>>(
        X, mask, W1, b1, W2, b2, W3, b3, out);
}